// DocumentGNN_1047972020879
// MI455X (gfx1250) — compile-verified
//
#include <hip/hip_runtime.h>

typedef __attribute__((ext_vector_type(2))) float v2f;
typedef __attribute__((ext_vector_type(8))) float v8f;

#define HID   64
#define FIN   22
#define COUT  6
#define BN_EPS 1e-5f

// ---------------------------------------------------------------------------
// Degree / normalization
// ---------------------------------------------------------------------------
__global__ __launch_bounds__(256) void k_init_deg(float* deg, int n) {
  int i = blockIdx.x * blockDim.x + threadIdx.x;
  if (i < n) deg[i] = 1.0f;                      // self-loop contribution
}

__global__ __launch_bounds__(256) void k_accum_deg(const int* __restrict__ dst,
                                                   float* deg, int e) {
  int i = blockIdx.x * blockDim.x + threadIdx.x;
  if (i < e) atomicAdd(&deg[dst[i]], 1.0f);
}

__global__ __launch_bounds__(256) void k_dinv(float* deg, int n) {
  int i = blockIdx.x * blockDim.x + threadIdx.x;
  if (i < n) deg[i] = rsqrtf(deg[i]);
}

// ---------------------------------------------------------------------------
// WMMA helper: D = A(16x4 f32) * B(4x16 f32) + C(16x16 f32)
// ---------------------------------------------------------------------------
__device__ __forceinline__ v8f wmma4(v2f a, v2f b, v8f c) {
  return __builtin_amdgcn_wmma_f32_16x16x4_f32(
      /*neg_a=*/false, a, /*neg_b=*/false, b,
      /*c_mod=*/(short)0, c, /*reuse_a=*/false, /*reuse_b=*/false);
}

// ---------------------------------------------------------------------------
// Embed: h = relu(x[N,22] @ W[22,64] + b) ; K padded 22 -> 24
// One wave computes a 16x64 tile: 4 accumulators of 16x16.
// ---------------------------------------------------------------------------
__global__ __launch_bounds__(256) void k_embed(const float* __restrict__ x,
                                               const float* __restrict__ W,
                                               const float* __restrict__ bias,
                                               float* __restrict__ h, int n) {
  int wid  = (blockIdx.x * blockDim.x + threadIdx.x) >> 5;
  int lane = threadIdx.x & 31;
  int row0 = wid * 16;
  if (row0 >= n) return;                       // wave-uniform guard (EXEC stays full)

  int m  = lane & 15;
  int kh = (lane >> 4) << 1;                   // lane half selects K-pair {0,1} or {2,3}
  int arow = row0 + m; if (arow > n - 1) arow = n - 1;

  v8f acc[4] = {};
  for (int k0 = 0; k0 < 24; k0 += 4) {
    int ka = k0 + kh;
    v2f a;
    a.x = (ka     < FIN) ? x[(size_t)arow * FIN + ka]     : 0.0f;
    a.y = (ka + 1 < FIN) ? x[(size_t)arow * FIN + ka + 1] : 0.0f;
#pragma unroll
    for (int t = 0; t < 4; ++t) {
      v2f b;
      b.x = (ka     < FIN) ? W[ka       * HID + t * 16 + m] : 0.0f;
      b.y = (ka + 1 < FIN) ? W[(ka + 1) * HID + t * 16 + m] : 0.0f;
      acc[t] = wmma4(a, b, acc[t]);
    }
  }
#pragma unroll
  for (int r = 0; r < 8; ++r) {
    int row = row0 + r + ((lane >> 4) << 3);   // C/D: lanes 16-31 hold M = r+8
    if (row < n) {
#pragma unroll
      for (int t = 0; t < 4; ++t) {
        int col = t * 16 + m;
        float v = acc[t][r] + bias[col];
        h[(size_t)row * HID + col] = v > 0.0f ? v : 0.0f;
      }
    }
  }
}

// ---------------------------------------------------------------------------
// GCN layer GEMM: t = h_in @ W (64x64), fused self-loop init: agg = t * dinv^2
// ---------------------------------------------------------------------------
__global__ __launch_bounds__(256) void k_gcn_gemm(const float* __restrict__ hin,
                                                  const float* __restrict__ W,
                                                  const float* __restrict__ dinv,
                                                  float* __restrict__ tout,
                                                  float* __restrict__ agg, int n) {
  int wid  = (blockIdx.x * blockDim.x + threadIdx.x) >> 5;
  int lane = threadIdx.x & 31;
  int row0 = wid * 16;
  if (row0 >= n) return;

  int m  = lane & 15;
  int kh = (lane >> 4) << 1;
  int arow = row0 + m; if (arow > n - 1) arow = n - 1;

  v8f acc[4] = {};
  for (int k0 = 0; k0 < HID; k0 += 4) {
    int ka = k0 + kh;
    v2f a;
    a.x = hin[(size_t)arow * HID + ka];
    a.y = hin[(size_t)arow * HID + ka + 1];
#pragma unroll
    for (int t = 0; t < 4; ++t) {
      v2f b;
      b.x = W[ka       * HID + t * 16 + m];
      b.y = W[(ka + 1) * HID + t * 16 + m];
      acc[t] = wmma4(a, b, acc[t]);
    }
  }
#pragma unroll
  for (int r = 0; r < 8; ++r) {
    int row = row0 + r + ((lane >> 4) << 3);
    if (row < n) {
      float di  = dinv[row];
      float di2 = di * di;
#pragma unroll
      for (int t = 0; t < 4; ++t) {
        int col = t * 16 + m;
        float v = acc[t][r];
        tout[(size_t)row * HID + col] = v;
        agg [(size_t)row * HID + col] = v * di2;   // self-loop term pre-seeded
      }
    }
  }
}

// ---------------------------------------------------------------------------
// Edge scatter: agg[dst] += dinv[src]*dinv[dst] * t[src]
// One wave per edge; each lane handles 2 features (coalesced 256B gather).
// ---------------------------------------------------------------------------
__global__ __launch_bounds__(256) void k_scatter(const int* __restrict__ src,
                                                 const int* __restrict__ dst,
                                                 const float* __restrict__ dinv,
                                                 const float* __restrict__ t,
                                                 float* agg, int e) {
  int gid  = blockIdx.x * blockDim.x + threadIdx.x;
  int edge = gid >> 5;
  if (edge >= e) return;
  int lane = gid & 31;

  int s = src[edge];
  int d = dst[edge];
  float coef = dinv[s] * dinv[d];

  int f = lane << 1;
  const float2 v = *reinterpret_cast<const float2*>(t + (size_t)s * HID + f);
  atomicAdd(agg + (size_t)d * HID + f,     coef * v.x);
  atomicAdd(agg + (size_t)d * HID + f + 1, coef * v.y);
}

// ---------------------------------------------------------------------------
// Fused bias + BatchNorm(eval) + ReLU
// ---------------------------------------------------------------------------
__global__ __launch_bounds__(256) void k_bn_relu(const float* __restrict__ agg,
                                                 const float* __restrict__ b,
                                                 const float* __restrict__ g,
                                                 const float* __restrict__ be,
                                                 const float* __restrict__ mean,
                                                 const float* __restrict__ var,
                                                 float* __restrict__ out, int total) {
  int i = blockIdx.x * blockDim.x + threadIdx.x;
  if (i >= total) return;
  int f = i & (HID - 1);
  float v = agg[i] + b[f];
  v = (v - mean[f]) * rsqrtf(var[f] + BN_EPS) * g[f] + be[f];
  out[i] = v > 0.0f ? v : 0.0f;
}

// ---------------------------------------------------------------------------
// Classifier: out = h @ W_cls[64,6] + b_cls  (C=6, plain VALU)
// ---------------------------------------------------------------------------
__global__ __launch_bounds__(256) void k_cls(const float* __restrict__ h,
                                             const float* __restrict__ W,
                                             const float* __restrict__ b,
                                             float* __restrict__ out, int n) {
  int i = blockIdx.x * blockDim.x + threadIdx.x;
  if (i >= n) return;
  float acc[COUT];
#pragma unroll
  for (int c = 0; c < COUT; ++c) acc[c] = b[c];
  for (int k = 0; k < HID; ++k) {
    float hv = h[(size_t)i * HID + k];
#pragma unroll
    for (int c = 0; c < COUT; ++c) acc[c] += hv * W[k * COUT + c];
  }
#pragma unroll
  for (int c = 0; c < COUT; ++c) out[(size_t)i * COUT + c] = acc[c];
}

// ---------------------------------------------------------------------------
extern "C" void kernel_launch(void* const* d_in, const int* in_sizes, int n_in,
                              void* d_out, int out_size, void* d_ws, size_t ws_size,
                              hipStream_t stream) {
  (void)n_in; (void)out_size; (void)ws_size;

  const float* x     = (const float*)d_in[0];
  const int*   ei    = (const int*)  d_in[1];
  const float* W_emb = (const float*)d_in[2];
  const float* b_emb = (const float*)d_in[3];
  const float* W1    = (const float*)d_in[4];
  const float* b1    = (const float*)d_in[5];
  const float* g1    = (const float*)d_in[6];
  const float* be1   = (const float*)d_in[7];
  const float* m1    = (const float*)d_in[8];
  const float* v1    = (const float*)d_in[9];
  const float* W2    = (const float*)d_in[10];
  const float* b2    = (const float*)d_in[11];
  const float* g2    = (const float*)d_in[12];
  const float* be2   = (const float*)d_in[13];
  const float* m2    = (const float*)d_in[14];
  const float* v2    = (const float*)d_in[15];
  const float* W_cls = (const float*)d_in[16];
  const float* b_cls = (const float*)d_in[17];

  const int n = in_sizes[0] / FIN;     // 100000
  const int e = in_sizes[1] / 2;       // 1200000
  const int* src = ei;
  const int* dst = ei + e;

  float* ws   = (float*)d_ws;
  float* dinv = ws;                          // [n]
  float* B0   = ws + n;                      // [n*HID]
  float* B1   = B0 + (size_t)n * HID;        // [n*HID]
  float* B2   = B1 + (size_t)n * HID;        // [n*HID]

  auto cdiv = [](long a, long b) { return (int)((a + b - 1) / b); };

  // normalization coefficients
  k_init_deg <<<cdiv(n, 256), 256, 0, stream>>>(dinv, n);
  k_accum_deg<<<cdiv(e, 256), 256, 0, stream>>>(dst, dinv, e);
  k_dinv     <<<cdiv(n, 256), 256, 0, stream>>>(dinv, n);

  const int gthreads = cdiv(n, 16) * 32;     // one wave per 16-row tile

  // embedding
  k_embed<<<cdiv(gthreads, 256), 256, 0, stream>>>(x, W_emb, b_emb, B0, n);

  // layer 1
  k_gcn_gemm<<<cdiv(gthreads, 256), 256, 0, stream>>>(B0, W1, dinv, B1, B2, n);
  k_scatter <<<cdiv((long)e * 32, 256), 256, 0, stream>>>(src, dst, dinv, B1, B2, e);
  k_bn_relu <<<cdiv((long)n * HID, 256), 256, 0, stream>>>(B2, b1, g1, be1, m1, v1, B0, n * HID);

  // layer 2
  k_gcn_gemm<<<cdiv(gthreads, 256), 256, 0, stream>>>(B0, W2, dinv, B1, B2, n);
  k_scatter <<<cdiv((long)e * 32, 256), 256, 0, stream>>>(src, dst, dinv, B1, B2, e);
  k_bn_relu <<<cdiv((long)n * HID, 256), 256, 0, stream>>>(B2, b2, g2, be2, m2, v2, B0, n * HID);

  // classifier
  k_cls<<<cdiv(n, 256), 256, 0, stream>>>(B0, W_cls, b_cls, (float*)d_out, n);
}